// DenseImplementation_11098195493543
// MI455X (gfx1250) — compile-verified
//
#include <hip/hip_runtime.h>
#include <hip/hip_bf16.h>

#define D_DIM 512
#define N1    5
#define K_DIM 2560
#define NB    32

typedef __attribute__((ext_vector_type(2))) float v2f;
typedef __attribute__((ext_vector_type(8))) float v8f;

// ---------------------------------------------------------------------------
// Generic WMMA GEMM (NT): C[M x N] = A[M x Kd] * B[N x Kd]^T   (mode 0)
//                         C       -= A * B^T                    (mode 1)
// One wave per 32x32 C tile. grid = (N/32, M/32), block = 32.
// ---------------------------------------------------------------------------
__global__ void __launch_bounds__(32) gemm_nt_wmma(
    const float* __restrict__ A, int lda,
    const float* __restrict__ B, int ldb,
    float* __restrict__ C, int ldc,
    int Kd, int mode) {
  const int lane = threadIdx.x & 31;
  const int half = lane >> 4;
  const int lrow = lane & 15;
  const int m0 = blockIdx.y << 5;
  const int n0 = blockIdx.x << 5;
  v8f c00 = {}, c01 = {}, c10 = {}, c11 = {};
  const float* Ar0 = A + (size_t)(m0 + lrow) * lda;
  const float* Ar1 = Ar0 + (size_t)16 * lda;
  const float* Br0 = B + (size_t)(n0 + lrow) * ldb;
  const float* Br1 = Br0 + (size_t)16 * ldb;
  const int koff = half << 1;
  for (int k = 0; k < Kd; k += 4) {
    const int kk = k + koff;
    v2f a0 = *(const v2f*)(Ar0 + kk);
    v2f a1 = *(const v2f*)(Ar1 + kk);
    v2f b0 = *(const v2f*)(Br0 + kk);
    v2f b1 = *(const v2f*)(Br1 + kk);
    c00 = __builtin_amdgcn_wmma_f32_16x16x4_f32(false, a0, false, b0, (short)0, c00, false, false);
    c01 = __builtin_amdgcn_wmma_f32_16x16x4_f32(false, a0, false, b1, (short)0, c01, false, false);
    c10 = __builtin_amdgcn_wmma_f32_16x16x4_f32(false, a1, false, b0, (short)0, c10, false, false);
    c11 = __builtin_amdgcn_wmma_f32_16x16x4_f32(false, a1, false, b1, (short)0, c11, false, false);
  }
  const int r0 = m0 + (half << 3);
  const int ca = n0 + lrow;
#pragma unroll
  for (int v = 0; v < 8; ++v) {
    size_t i00 = (size_t)(r0 + v) * ldc + ca;
    size_t i10 = (size_t)(r0 + 16 + v) * ldc + ca;
    if (mode == 0) {
      C[i00]      = c00[v]; C[i00 + 16] = c01[v];
      C[i10]      = c10[v]; C[i10 + 16] = c11[v];
    } else {
      C[i00]      -= c00[v]; C[i00 + 16] -= c01[v];
      C[i10]      -= c10[v]; C[i10 + 16] -= c11[v];
    }
  }
}

// ---------------------------------------------------------------------------
// Big-tile WMMA GEMM (NT): 64x64 C tile per wave, 16 accumulators.
// grid = (N/64, M/64), block = 32. Kd % 16 == 0.
// 16 WMMAs (32768 flops) per 2KB of fragment traffic per k-step group.
// ---------------------------------------------------------------------------
__global__ void __launch_bounds__(32) gemm_nt64_wmma(
    const float* __restrict__ A, int lda,
    const float* __restrict__ B, int ldb,
    float* __restrict__ C, int ldc,
    int Kd, int mode) {
  const int lane = threadIdx.x & 31;
  const int half = lane >> 4;
  const int lrow = lane & 15;
  const int m0 = blockIdx.y << 6;
  const int n0 = blockIdx.x << 6;
  v8f acc[4][4];
#pragma unroll
  for (int i = 0; i < 4; ++i)
#pragma unroll
    for (int j = 0; j < 4; ++j) acc[i][j] = {};
  const float* Ar[4];
  const float* Br[4];
#pragma unroll
  for (int i = 0; i < 4; ++i) {
    Ar[i] = A + (size_t)(m0 + 16 * i + lrow) * lda;
    Br[i] = B + (size_t)(n0 + 16 * i + lrow) * ldb;
  }
  const int koff = half << 1;
  for (int k = 0; k < Kd; k += 16) {
#pragma unroll
    for (int i = 0; i < 4; ++i) {
      __builtin_prefetch(Ar[i] + k + 80, 0, 0);
      __builtin_prefetch(Br[i] + k + 80, 0, 0);
    }
#pragma unroll
    for (int k4 = 0; k4 < 16; k4 += 4) {
      const int kk = k + k4 + koff;
      v2f af[4], bf[4];
#pragma unroll
      for (int i = 0; i < 4; ++i) {
        af[i] = *(const v2f*)(Ar[i] + kk);
        bf[i] = *(const v2f*)(Br[i] + kk);
      }
#pragma unroll
      for (int i = 0; i < 4; ++i)
#pragma unroll
        for (int j = 0; j < 4; ++j)
          acc[i][j] = __builtin_amdgcn_wmma_f32_16x16x4_f32(
              false, af[i], false, bf[j], (short)0, acc[i][j], false, false);
    }
  }
#pragma unroll
  for (int i = 0; i < 4; ++i) {
    const int r0 = m0 + 16 * i + (half << 3);
#pragma unroll
    for (int j = 0; j < 4; ++j) {
      const int ca = n0 + 16 * j + lrow;
#pragma unroll
      for (int v = 0; v < 8; ++v) {
        size_t idx = (size_t)(r0 + v) * ldc + ca;
        if (mode == 0) C[idx] = acc[i][j][v];
        else           C[idx] -= acc[i][j][v];
      }
    }
  }
}

// ---------------------------------------------------------------------------
// Generic WMMA GEMM (NN): C = A*B (0) ; C -= A*B (1) ; C = Dsrc - A*B (2)
// One wave per 32x32 tile.
// ---------------------------------------------------------------------------
__global__ void __launch_bounds__(32) gemm_nn_wmma(
    const float* __restrict__ A, int lda,
    const float* __restrict__ B, int ldb,
    float* __restrict__ C, int ldc,
    const float* __restrict__ Dsrc, int ldd,
    int Kd, int mode) {
  const int lane = threadIdx.x & 31;
  const int half = lane >> 4;
  const int lrow = lane & 15;
  const int m0 = blockIdx.y << 5;
  const int n0 = blockIdx.x << 5;
  v8f c00 = {}, c01 = {}, c10 = {}, c11 = {};
  const float* Ar0 = A + (size_t)(m0 + lrow) * lda;
  const float* Ar1 = Ar0 + (size_t)16 * lda;
  const float* Bc  = B + n0 + lrow;
  const int koff = half << 1;
  for (int k = 0; k < Kd; k += 4) {
    const int kk = k + koff;
    v2f a0 = *(const v2f*)(Ar0 + kk);
    v2f a1 = *(const v2f*)(Ar1 + kk);
    v2f b0, b1;
    b0.x = Bc[(size_t)kk * ldb];
    b0.y = Bc[(size_t)(kk + 1) * ldb];
    b1.x = Bc[(size_t)kk * ldb + 16];
    b1.y = Bc[(size_t)(kk + 1) * ldb + 16];
    c00 = __builtin_amdgcn_wmma_f32_16x16x4_f32(false, a0, false, b0, (short)0, c00, false, false);
    c01 = __builtin_amdgcn_wmma_f32_16x16x4_f32(false, a0, false, b1, (short)0, c01, false, false);
    c10 = __builtin_amdgcn_wmma_f32_16x16x4_f32(false, a1, false, b0, (short)0, c10, false, false);
    c11 = __builtin_amdgcn_wmma_f32_16x16x4_f32(false, a1, false, b1, (short)0, c11, false, false);
  }
  const int r0 = m0 + (half << 3);
  const int ca = n0 + lrow;
#pragma unroll
  for (int v = 0; v < 8; ++v) {
    size_t i00 = (size_t)(r0 + v) * ldc + ca;
    size_t i10 = (size_t)(r0 + 16 + v) * ldc + ca;
    if (mode == 0) {
      C[i00]      = c00[v]; C[i00 + 16] = c01[v];
      C[i10]      = c10[v]; C[i10 + 16] = c11[v];
    } else if (mode == 1) {
      C[i00]      -= c00[v]; C[i00 + 16] -= c01[v];
      C[i10]      -= c10[v]; C[i10 + 16] -= c11[v];
    } else {
      size_t j00 = (size_t)(r0 + v) * ldd + ca;
      size_t j10 = (size_t)(r0 + 16 + v) * ldd + ca;
      C[i00]      = Dsrc[j00]      - c00[v];
      C[i00 + 16] = Dsrc[j00 + 16] - c01[v];
      C[i10]      = Dsrc[j10]      - c10[v];
      C[i10 + 16] = Dsrc[j10 + 16] - c11[v];
    }
  }
}

// ---------------------------------------------------------------------------
// Big-tile WMMA GEMM (NN): 64x64 C tile per wave. Kd % 16 == 0.
// ---------------------------------------------------------------------------
__global__ void __launch_bounds__(32) gemm_nn64_wmma(
    const float* __restrict__ A, int lda,
    const float* __restrict__ B, int ldb,
    float* __restrict__ C, int ldc,
    const float* __restrict__ Dsrc, int ldd,
    int Kd, int mode) {
  const int lane = threadIdx.x & 31;
  const int half = lane >> 4;
  const int lrow = lane & 15;
  const int m0 = blockIdx.y << 6;
  const int n0 = blockIdx.x << 6;
  v8f acc[4][4];
#pragma unroll
  for (int i = 0; i < 4; ++i)
#pragma unroll
    for (int j = 0; j < 4; ++j) acc[i][j] = {};
  const float* Ar[4];
#pragma unroll
  for (int i = 0; i < 4; ++i)
    Ar[i] = A + (size_t)(m0 + 16 * i + lrow) * lda;
  const float* Bc = B + n0 + lrow;
  const int koff = half << 1;
  for (int k = 0; k < Kd; k += 16) {
#pragma unroll
    for (int i = 0; i < 4; ++i) __builtin_prefetch(Ar[i] + k + 80, 0, 0);
#pragma unroll
    for (int k4 = 0; k4 < 16; k4 += 4) {
      const int kk = k + k4 + koff;
      v2f af[4], bf[4];
#pragma unroll
      for (int i = 0; i < 4; ++i) af[i] = *(const v2f*)(Ar[i] + kk);
      const float* brow0 = Bc + (size_t)kk * ldb;
      const float* brow1 = Bc + (size_t)(kk + 1) * ldb;
#pragma unroll
      for (int j = 0; j < 4; ++j) {
        bf[j].x = brow0[16 * j];
        bf[j].y = brow1[16 * j];
      }
#pragma unroll
      for (int i = 0; i < 4; ++i)
#pragma unroll
        for (int j = 0; j < 4; ++j)
          acc[i][j] = __builtin_amdgcn_wmma_f32_16x16x4_f32(
              false, af[i], false, bf[j], (short)0, acc[i][j], false, false);
    }
  }
#pragma unroll
  for (int i = 0; i < 4; ++i) {
    const int r0 = m0 + 16 * i + (half << 3);
#pragma unroll
    for (int j = 0; j < 4; ++j) {
      const int ca = n0 + 16 * j + lrow;
#pragma unroll
      for (int v = 0; v < 8; ++v) {
        size_t idx = (size_t)(r0 + v) * ldc + ca;
        if (mode == 0)      C[idx] = acc[i][j][v];
        else if (mode == 1) C[idx] -= acc[i][j][v];
        else                C[idx] = Dsrc[(size_t)(r0 + v) * ldd + ca] - acc[i][j][v];
      }
    }
  }
}

// ---------------------------------------------------------------------------
// 32x32 in-place Cholesky of diagonal block (lower). 1 block x 32 threads.
// ---------------------------------------------------------------------------
__global__ void __launch_bounds__(32) potf2_32(float* A, int ld, int off) {
  __shared__ float s[32][33];
  int t = threadIdx.x;
  for (int j = 0; j < 32; ++j) s[t][j] = A[(size_t)(off + t) * ld + off + j];
  __syncthreads();
  for (int j = 0; j < 32; ++j) {
    if (t == j) s[j][j] = sqrtf(s[j][j]);
    __syncthreads();
    if (t > j) s[t][j] /= s[j][j];
    __syncthreads();
    if (t > j) {
      for (int c = j + 1; c <= t; ++c) s[t][c] -= s[t][j] * s[c][j];
    }
    __syncthreads();
  }
  for (int j = 0; j <= t; ++j) A[(size_t)(off + t) * ld + off + j] = s[t][j];
}

// Y[r, 0:32] <- Y[r, 0:32] * L^{-T}  (forward substitution per row).
__global__ void trsm_fwd(float* Y, int ldy, int nrows,
                         const float* __restrict__ Lsrc, int ldl) {
  __shared__ float Lb[32][33];
  int tid = threadIdx.x;
  for (int e = tid; e < 1024; e += blockDim.x) {
    int i = e >> 5, j = e & 31;
    Lb[i][j] = Lsrc[(size_t)i * ldl + j];
  }
  __syncthreads();
  int r = blockIdx.x * blockDim.x + tid;
  if (r >= nrows) return;
  float* row = Y + (size_t)r * ldy;
  float x[32];
  for (int j = 0; j < 32; ++j) {
    float s = row[j];
    for (int i = 0; i < j; ++i) s -= x[i] * Lb[j][i];
    x[j] = s / Lb[j][j];
  }
  for (int j = 0; j < 32; ++j) row[j] = x[j];
}

// Y[r, 0:32] <- Y[r, 0:32] * L^{-1}  (backward substitution per row).
__global__ void trsm_bwd(float* Y, int ldy, int nrows,
                         const float* __restrict__ Lsrc, int ldl) {
  __shared__ float Lb[32][33];
  int tid = threadIdx.x;
  for (int e = tid; e < 1024; e += blockDim.x) {
    int i = e >> 5, j = e & 31;
    Lb[i][j] = Lsrc[(size_t)i * ldl + j];
  }
  __syncthreads();
  int r = blockIdx.x * blockDim.x + tid;
  if (r >= nrows) return;
  float* row = Y + (size_t)r * ldy;
  float x[32];
  for (int j = 31; j >= 0; --j) {
    float s = row[j];
    for (int i = j + 1; i < 32; ++i) s -= x[i] * Lb[i][j];
    x[j] = s / Lb[j][j];
  }
  for (int j = 0; j < 32; ++j) row[j] = x[j];
}

// ---------------------------------------------------------------------------
// Small / elementwise kernels
// ---------------------------------------------------------------------------
__global__ void prep_p(const float* __restrict__ dt, float* p, float* pinv) {
  int t = blockIdx.x * blockDim.x + threadIdx.x;
  if (t >= K_DIM) return;
  float adt = fabsf(dt[0]);
  int i = t % N1;                    // powers = [4,3,2,1,0]
  float pwv = (float)(N1 - 1 - i) + 0.5f;
  const float scales[N1] = {24.f, 6.f, 2.f, 1.f, 1.f};
  float sc = scales[i];
  float ppow = powf(adt, pwv);
  p[t] = ppow / sc;
  pinv[t] = sc / ppow;
}

__global__ void mext_kernel(const float* __restrict__ a, const float* __restrict__ m0v,
                            const float* __restrict__ p, const float* __restrict__ pinv,
                            float* mext) {
  int t = blockIdx.x * blockDim.x + threadIdx.x;
  if (t >= K_DIM) return;
  int b5 = (t / N1) * N1;
  float s = 0.f;
  for (int j = 0; j < N1; ++j)
    s += a[(size_t)t * K_DIM + b5 + j] * pinv[b5 + j] * m0v[b5 + j];
  mext[t] = p[t] * s;
}

__global__ void rowdot_sub(const float* __restrict__ M, const float* __restrict__ v,
                           const float* __restrict__ bias, float* out, int cols) {
  __shared__ float red[256];
  int r = blockIdx.x;
  float s = 0.f;
  for (int c = threadIdx.x; c < cols; c += blockDim.x)
    s += M[(size_t)r * cols + c] * v[c];
  red[threadIdx.x] = s; __syncthreads();
  for (int st = 128; st > 0; st >>= 1) {
    if (threadIdx.x < st) red[threadIdx.x] += red[threadIdx.x + st];
    __syncthreads();
  }
  if (threadIdx.x == 0) out[r] = red[0] - bias[r];
}

__global__ void build_E(const float* __restrict__ H, const float* __restrict__ pinv,
                        const float* __restrict__ q, float* E) {
  int idx = blockIdx.x * blockDim.x + threadIdx.x;
  if (idx >= D_DIM * K_DIM) return;
  int r = idx / K_DIM;
  int t = idx % K_DIM;
  int b5 = (t / N1) * N1;
  int j = t % N1;
  float s = 0.f;
  for (int i = 0; i < N1; ++i)
    s += H[(size_t)r * K_DIM + b5 + i] * pinv[b5 + i] * q[(size_t)(b5 + i) * K_DIM + b5 + j];
  E[(size_t)r * K_DIM + t] = s;
}

__global__ void build_X(const float* __restrict__ a, const float* __restrict__ pinv,
                        const float* __restrict__ l0, float* X) {
  size_t idx = (size_t)blockIdx.x * blockDim.x + threadIdx.x;
  if (idx >= (size_t)K_DIM * K_DIM) return;
  int t = (int)(idx / K_DIM);
  int c = (int)(idx % K_DIM);
  int b5 = (t / N1) * N1;
  float s = 0.f;
  for (int j = 0; j < N1; ++j)
    s += a[(size_t)t * K_DIM + b5 + j] * pinv[b5 + j] * l0[(size_t)(b5 + j) * K_DIM + c];
  X[idx] = s;
}

__global__ void add_qq(const float* __restrict__ q, const float* __restrict__ diff, float* G) {
  int e = blockIdx.x * blockDim.x + threadIdx.x;
  if (e >= D_DIM * N1 * N1) return;
  int b = e / (N1 * N1);
  int ij = e % (N1 * N1);
  int i = ij / N1, j = ij % N1;
  int b5 = b * N1;
  float dv = diff[0];
  float s = 0.f;
  for (int k = 0; k < N1; ++k)
    s += q[(size_t)(b5 + i) * K_DIM + b5 + k] * q[(size_t)(b5 + j) * K_DIM + b5 + k];
  G[(size_t)(b5 + i) * K_DIM + b5 + j] += dv * dv * s;
}

__global__ void make_lext(const float* __restrict__ G, const float* __restrict__ p, float* LEXT) {
  size_t idx = (size_t)blockIdx.x * blockDim.x + threadIdx.x;
  if (idx >= (size_t)K_DIM * K_DIM) return;
  int i = (int)(idx / K_DIM);
  int j = (int)(idx % K_DIM);
  LEXT[idx] = (j <= i) ? p[i] * G[idx] : 0.f;
}

__global__ void solve_white(const float* __restrict__ L, const float* __restrict__ b,
                            float* diff_ws, float* diff_out) {
  __shared__ float xs[D_DIM];
  __shared__ float red[256];
  int t = threadIdx.x;
  for (int j = D_DIM - 1; j >= 0; --j) {
    float s = 0.f;
    for (int i = j + 1 + t; i < D_DIM; i += 256)
      s += L[(size_t)i * D_DIM + j] * xs[i];
    red[t] = s; __syncthreads();
    for (int st = 128; st > 0; st >>= 1) {
      if (t < st) red[t] += red[t + st];
      __syncthreads();
    }
    if (t == 0) xs[j] = (b[j] - red[0]) / L[(size_t)j * D_DIM + j];
    __syncthreads();
  }
  float s = xs[t] * xs[t] + xs[t + 256] * xs[t + 256];
  red[t] = s; __syncthreads();
  for (int st = 128; st > 0; st >>= 1) {
    if (t < st) red[t] += red[t + st];
    __syncthreads();
  }
  if (t == 0) {
    float dv = sqrtf(red[0] / (float)D_DIM);
    diff_ws[0] = dv;
    diff_out[0] = dv;
  }
}

__global__ void error_kernel(const float* __restrict__ L, const float* __restrict__ diff,
                             float* out) {
  __shared__ float red[128];
  int n = blockIdx.x;
  float s = 0.f;
  for (int j = threadIdx.x; j <= n; j += blockDim.x) {
    float v = L[(size_t)n * D_DIM + j];
    s += v * v;
  }
  red[threadIdx.x] = s; __syncthreads();
  for (int st = 64; st > 0; st >>= 1) {
    if (threadIdx.x < st) red[threadIdx.x] += red[threadIdx.x + st];
    __syncthreads();
  }
  if (threadIdx.x == 0) out[n] = diff[0] * sqrtf(red[0]);
}

__global__ void mcor_kernel(const float* __restrict__ gain, const float* __restrict__ mobs,
                            const float* __restrict__ mext, float* out) {
  __shared__ float red[128];
  int r = blockIdx.x;
  float s = 0.f;
  for (int c = threadIdx.x; c < D_DIM; c += blockDim.x)
    s += gain[(size_t)r * D_DIM + c] * mobs[c];
  red[threadIdx.x] = s; __syncthreads();
  for (int st = 64; st > 0; st >>= 1) {
    if (threadIdx.x < st) red[threadIdx.x] += red[threadIdx.x + st];
    __syncthreads();
  }
  if (threadIdx.x == 0) out[r] = mext[r] - red[0];
}

__global__ void u_kernel(const float* __restrict__ mcor, float* u_out) {
  int j = blockIdx.x * blockDim.x + threadIdx.x;
  if (j < D_DIM) u_out[j] = mcor[(size_t)N1 * j];
}

// ---------------------------------------------------------------------------
extern "C" void kernel_launch(void* const* d_in, const int* in_sizes, int n_in,
                              void* d_out, int out_size, void* d_ws, size_t ws_size,
                              hipStream_t stream) {
  const float* m0   = (const float*)d_in[0];
  const float* l0   = (const float*)d_in[1];
  const float* H    = (const float*)d_in[2];
  const float* bias = (const float*)d_in[3];
  const float* dt   = (const float*)d_in[4];
  const float* a    = (const float*)d_in[5];
  const float* q    = (const float*)d_in[6];
  float* out = (float*)d_out;
  float* ws  = (float*)d_ws;

  // workspace layout (floats)
  float* p    = ws;                       // 2560
  float* pinv = ws + 2560;                // 2560
  float* mext = ws + 5120;                // 2560
  float* mobs = ws + 7680;                // 512
  float* diff = ws + 8192;                // 1 (+pad)
  float* E    = ws + 8448;                // 512*2560  (reused as LON)
  float* G1   = E + (size_t)D_DIM * K_DIM;        // 512*512 (reused as G2 = l_obs)
  float* X    = G1 + (size_t)D_DIM * D_DIM;       // 2560*2560 (reused as LEXT)
  float* G    = X + (size_t)K_DIM * K_DIM;        // 2560*2560
  float* CC   = G + (size_t)K_DIM * K_DIM;        // 2560*512 (crosscov -> gain)

  // output layout: m_cor | l_cor | u | error | diffusion
  float* mcor_out = out;
  float* lcor_out = out + K_DIM;
  float* u_out    = lcor_out + (size_t)K_DIM * K_DIM;
  float* err_out  = u_out + D_DIM;
  float* diff_out = err_out + D_DIM;

  auto launch_nt = [&](const float* A, int lda, const float* B, int ldb,
                       float* C, int ldc, int M, int N, int Kd, int mode) {
    if ((M % 64 == 0) && (N % 64 == 0) && (Kd % 16 == 0))
      gemm_nt64_wmma<<<dim3(N / 64, M / 64), 32, 0, stream>>>(A, lda, B, ldb, C, ldc, Kd, mode);
    else
      gemm_nt_wmma<<<dim3(N / 32, M / 32), 32, 0, stream>>>(A, lda, B, ldb, C, ldc, Kd, mode);
  };
  auto launch_nn = [&](const float* A, int lda, const float* B, int ldb,
                       float* C, int ldc, const float* Dsrc, int ldd,
                       int M, int N, int Kd, int mode) {
    if ((M % 64 == 0) && (N % 64 == 0) && (Kd % 16 == 0))
      gemm_nn64_wmma<<<dim3(N / 64, M / 64), 32, 0, stream>>>(A, lda, B, ldb, C, ldc, Dsrc, ldd, Kd, mode);
    else
      gemm_nn_wmma<<<dim3(N / 32, M / 32), 32, 0, stream>>>(A, lda, B, ldb, C, ldc, Dsrc, ldd, Kd, mode);
  };

  auto chol = [&](float* Am, int n) {
    for (int off = 0; off < n; off += NB) {
      potf2_32<<<1, 32, 0, stream>>>(Am, n, off);
      int rem = n - off - NB;
      if (rem > 0) {
        trsm_fwd<<<(rem + 127) / 128, 128, 0, stream>>>(
            Am + (size_t)(off + NB) * n + off, n, rem,
            Am + (size_t)off * n + off, n);
        launch_nt(Am + (size_t)(off + NB) * n + off, n,
                  Am + (size_t)(off + NB) * n + off, n,
                  Am + (size_t)(off + NB) * n + (off + NB), n, rem, rem, NB, 1);
      }
    }
  };

  // 1. p, pinv, m_ext
  prep_p<<<(K_DIM + 255) / 256, 256, 0, stream>>>(dt, p, pinv);
  mext_kernel<<<(K_DIM + 255) / 256, 256, 0, stream>>>(a, m0, p, pinv, mext);
  // 2. m_obs = H @ m_ext - bias
  rowdot_sub<<<D_DIM, 256, 0, stream>>>(H, mext, bias, mobs, K_DIM);
  // 3. E = H @ (pinv * q)   (block-diagonal exploitation)
  build_E<<<(D_DIM * K_DIM + 255) / 256, 256, 0, stream>>>(H, pinv, q, E);
  // 4. G1 = E E^T  -> Cholesky -> l_obs_raw
  launch_nt(E, K_DIM, E, K_DIM, G1, D_DIM, D_DIM, D_DIM, K_DIM, 0);
  chol(G1, D_DIM);
  // 5. res_white solve, diffusion, error
  solve_white<<<1, 256, 0, stream>>>(G1, mobs, diff, diff_out);
  error_kernel<<<D_DIM, 128, 0, stream>>>(G1, diff, err_out);
  // 6. X = a @ (pinv * l0)
  build_X<<<(int)(((size_t)K_DIM * K_DIM + 255) / 256), 256, 0, stream>>>(a, pinv, l0, X);
  // 7. G = X X^T + diff^2 * q q^T  -> Cholesky -> l_ext_p
  launch_nt(X, K_DIM, X, K_DIM, G, K_DIM, K_DIM, K_DIM, K_DIM, 0);
  add_qq<<<(D_DIM * N1 * N1 + 255) / 256, 256, 0, stream>>>(q, diff, G);
  chol(G, K_DIM);
  // 8. LEXT = p[:,None] * tril(G)   (into X buffer)
  make_lext<<<(int)(((size_t)K_DIM * K_DIM + 255) / 256), 256, 0, stream>>>(G, p, X);
  // 9. LON = H @ LEXT  (into E buffer)
  launch_nn(H, K_DIM, X, K_DIM, E, K_DIM, nullptr, 0, D_DIM, K_DIM, K_DIM, 0);
  // 10. G2 = LON LON^T -> Cholesky -> l_obs  (into G1 buffer)
  launch_nt(E, K_DIM, E, K_DIM, G1, D_DIM, D_DIM, D_DIM, K_DIM, 0);
  chol(G1, D_DIM);
  // 11. CC = LEXT @ LON^T  (crosscov)
  launch_nt(X, K_DIM, E, K_DIM, CC, D_DIM, K_DIM, D_DIM, K_DIM, 0);
  // 12. gain = CC * L^{-T} * L^{-1}  (blocked TRSM, in place in CC)
  for (int jb = 0; jb < D_DIM; jb += NB) {
    trsm_fwd<<<(K_DIM + 127) / 128, 128, 0, stream>>>(
        CC + jb, D_DIM, K_DIM, G1 + (size_t)jb * D_DIM + jb, D_DIM);
    int rem = D_DIM - jb - NB;
    if (rem > 0)
      launch_nt(CC + jb, D_DIM, G1 + (size_t)(jb + NB) * D_DIM + jb, D_DIM,
                CC + jb + NB, D_DIM, K_DIM, rem, NB, 1);
  }
  for (int jb = D_DIM - NB; jb >= 0; jb -= NB) {
    trsm_bwd<<<(K_DIM + 127) / 128, 128, 0, stream>>>(
        CC + jb, D_DIM, K_DIM, G1 + (size_t)jb * D_DIM + jb, D_DIM);
    if (jb > 0)
      launch_nn(CC + jb, D_DIM, G1 + (size_t)jb * D_DIM, D_DIM, CC, D_DIM,
                nullptr, 0, K_DIM, jb, NB, 1);
  }
  // 13. m_cor = m_ext - gain @ m_obs ; u
  mcor_kernel<<<K_DIM, 128, 0, stream>>>(CC, mobs, mext, mcor_out);
  u_kernel<<<2, 256, 0, stream>>>(mcor_out, u_out);
  // 14. l_cor = LEXT - gain @ LON
  launch_nn(CC, D_DIM, E, K_DIM, lcor_out, K_DIM, X, K_DIM, K_DIM, K_DIM, D_DIM, 2);
}